// ReChannel_normal_conv_90108413870131
// MI455X (gfx1250) — compile-verified
//
#include <hip/hip_runtime.h>
#include <math.h>

typedef __attribute__((ext_vector_type(2))) float v2f;
typedef __attribute__((ext_vector_type(4))) float v4f;
typedef __attribute__((ext_vector_type(8))) float v8f;

#define B_DIM 16
#define C_DIM 2048
#define HW    4096      // 64*64
#define K_DIM 256       // C / 8
#define KR_DIM 128      // K / r
#define ORI   8

// ---------------------------------------------------------------------------
// Kernel 1: global avg + max pool over H*W=4096 for each (b,c).
// One 256-thread block per (b,c); streaming float4 (b128) non-temporal loads.
// ---------------------------------------------------------------------------
__global__ __launch_bounds__(256) void pool_kernel(const float* __restrict__ x,
                                                   float* __restrict__ avg_out,
                                                   float* __restrict__ max_out) {
  const int bc = blockIdx.x;
  const v4f* p = (const v4f*)(x + (size_t)bc * HW);   // 1024 float4 per (b,c)
  const int t = threadIdx.x;

  float s  = 0.0f;
  float mx = -INFINITY;
#pragma unroll
  for (int i = 0; i < 4; ++i) {
    v4f v = __builtin_nontemporal_load(&p[t + i * 256]);
    s += (v.x + v.y) + (v.z + v.w);
    mx = fmaxf(mx, fmaxf(fmaxf(v.x, v.y), fmaxf(v.z, v.w)));
  }

  // wave32 butterfly reduction
#pragma unroll
  for (int off = 16; off >= 1; off >>= 1) {
    s += __shfl_xor(s, off, 32);
    mx = fmaxf(mx, __shfl_xor(mx, off, 32));
  }

  __shared__ float ls[8];
  __shared__ float lm[8];
  const int wave = t >> 5;
  const int lane = t & 31;
  if (lane == 0) { ls[wave] = s; lm[wave] = mx; }
  __syncthreads();
  if (t == 0) {
    float ts = 0.0f, tm = -INFINITY;
#pragma unroll
    for (int i = 0; i < 8; ++i) { ts += ls[i]; tm = fmaxf(tm, lm[i]); }
    avg_out[bc] = ts * (1.0f / (float)HW);
    max_out[bc] = tm;
  }
}

// ---------------------------------------------------------------------------
// Kernel 2: SE branch with V_WMMA_F32_16X16X4_F32.
//   F[br][16][256] = channel conv of pooled (br0=avg, br1=max)
//   H[br] = relu(F[br] @ W1^T)          (GEMM1, per branch)
//   Y     = (H[0]+H[1]) @ W2^T          (GEMM2 is linear -> sum branches first)
//   out   = sigmoid(Y) repeated over 8 orientations
// Single block, 256 threads = 8 waves.
// ---------------------------------------------------------------------------
__global__ __launch_bounds__(256) void se_kernel(const float* __restrict__ pooled, // [2][16][2048]
                                                 const float* __restrict__ w_ch,   // [8]
                                                 const float* __restrict__ w1,     // [128][256]
                                                 const float* __restrict__ w2,     // [256][128]
                                                 float* __restrict__ out) {        // [16][2048]
  __shared__ float Fs[2][B_DIM][K_DIM];    // 32 KB
  __shared__ float Hs[2][B_DIM][KR_DIM];   // 16 KB
  const int tid = threadIdx.x;

  float wc[8];
#pragma unroll
  for (int o = 0; o < 8; ++o) wc[o] = w_ch[o];

  // ---- Step A: F[br][b][k] = sum_o pooled[br][b][k*8+o] * w_ch[o] ----
  for (int e = tid; e < 2 * B_DIM * K_DIM; e += 256) {
    const int br = e >> 12;
    const int b  = (e >> 8) & 15;
    const int k  = e & 255;
    const float* pp = pooled + (size_t)br * (B_DIM * C_DIM) + b * C_DIM + k * ORI;
    float f = 0.0f;
#pragma unroll
    for (int o = 0; o < 8; ++o) f += pp[o] * wc[o];
    Fs[br][b][k] = f;
  }
  __syncthreads();

  const int wave  = tid >> 5;
  const int lane  = tid & 31;
  const int mrow  = lane & 15;          // A-frag row (M)
  const int ncol  = lane & 15;          // B-frag col (N)
  const int khalf = (lane >> 4) * 2;    // lanes 0-15 -> K0,1 ; lanes 16-31 -> K2,3

  // ---- Step B: H[br] = relu(F[br] @ W1^T); wave w owns N-tile jt=w ----
  const int jt = wave;
#pragma unroll
  for (int br = 0; br < 2; ++br) {
    v8f acc = {};
    for (int kk = 0; kk < K_DIM / 4; ++kk) {
      const int kb = kk * 4 + khalf;
      v2f a = { Fs[br][mrow][kb], Fs[br][mrow][kb + 1] };
      const float* wp = w1 + (size_t)(jt * 16 + ncol) * K_DIM + kb; // W1^T[k, n] = w1[n*256+k]
      v2f bfrag = { wp[0], wp[1] };
      acc = __builtin_amdgcn_wmma_f32_16x16x4_f32(false, a, false, bfrag,
                                                  (short)0, acc, false, false);
    }
#pragma unroll
    for (int r = 0; r < 8; ++r) {
      const int m = r + ((lane >> 4) << 3);   // C/D layout: hi-half lanes -> M+8
      Hs[br][m][jt * 16 + ncol] = fmaxf(acc[r], 0.0f);
    }
  }
  __syncthreads();

  // ---- Step C: Y = (H0+H1) @ W2^T ; wave w owns N-tiles {w, w+8} ----
#pragma unroll
  for (int tt = 0; tt < 2; ++tt) {
    const int nt = wave + tt * 8;
    v8f acc = {};
    for (int kk = 0; kk < KR_DIM / 4; ++kk) {
      const int kb = kk * 4 + khalf;
      v2f a = { Hs[0][mrow][kb] + Hs[1][mrow][kb],
                Hs[0][mrow][kb + 1] + Hs[1][mrow][kb + 1] };
      const float* wp = w2 + (size_t)(nt * 16 + ncol) * KR_DIM + kb; // W2^T[j, k] = w2[k*128+j]
      v2f bfrag = { wp[0], wp[1] };
      acc = __builtin_amdgcn_wmma_f32_16x16x4_f32(false, a, false, bfrag,
                                                  (short)0, acc, false, false);
    }
#pragma unroll
    for (int r = 0; r < 8; ++r) {
      const int m    = r + ((lane >> 4) << 3);
      const int kout = nt * 16 + ncol;
      const float y  = 1.0f / (1.0f + __expf(-acc[r]));
      v4f yv = { y, y, y, y };
      v4f* op = (v4f*)(out + (size_t)m * C_DIM + kout * ORI);
      op[0] = yv;   // orientations 0..3
      op[1] = yv;   // orientations 4..7
    }
  }
}

// ---------------------------------------------------------------------------
extern "C" void kernel_launch(void* const* d_in, const int* in_sizes, int n_in,
                              void* d_out, int out_size, void* d_ws, size_t ws_size,
                              hipStream_t stream) {
  const float* x    = (const float*)d_in[0];  // [16,2048,64,64]
  const float* w_ch = (const float*)d_in[1];  // [8]
  const float* w1   = (const float*)d_in[2];  // [128,256]
  const float* w2   = (const float*)d_in[3];  // [256,128]
  float* out = (float*)d_out;                 // [16,2048] (att, squeezed)

  float* ws  = (float*)d_ws;
  float* avg = ws;                            // [16*2048] floats
  float* mx  = ws + B_DIM * C_DIM;            // [16*2048] floats  (256 KB total)

  pool_kernel<<<B_DIM * C_DIM, 256, 0, stream>>>(x, avg, mx);
  se_kernel<<<1, 256, 0, stream>>>(avg, w_ch, w1, w2, out);
}